// TensorBoard_4423816315112
// MI455X (gfx1250) — compile-verified
//
#include <hip/hip_runtime.h>
#include <stdint.h>

// Super-ko Zobrist delta: out[b,i] = hash[b] ^ place_delta[b,i] ^ (keep ? seg_xor : 0)
// Memory-bound integer kernel; CDNA5 features used:
//   - global_load_async_to_lds_b32 (ASYNCcnt) to stage the 4.3KB Zobrist table in LDS
//   - s_wait_asynccnt 0 before the workgroup barrier
//   - global_prefetch_b8 on each thread's CSR segment
// wave32: 256-thread blocks.

#define N2 361
#define NZ (3 * N2) /* 1083 ints = 4332 bytes */

__global__ __launch_bounds__(256) void superko_hash_kernel(
    const uint8_t* __restrict__ legal_mask,       // (B,N2) bool (1 byte)
    const int*     __restrict__ current_player,   // (B,)
    const int*     __restrict__ current_hash,     // (B,)
    const int*     __restrict__ ZposT,            // (3,N2)
    const uint8_t* __restrict__ can_capture_any,  // (B,N2) bool (1 byte)
    const int*     __restrict__ cap_indptr,       // (B*N2+1,)
    const int*     __restrict__ cap_indices,      // (L,)
    int*           __restrict__ out,              // (B,N2)
    int total)                                    // B*N2
{
    __shared__ int zs[NZ];

    // Stage ZposT into LDS with gfx1250 async global->LDS copies (per-lane,
    // EXEC-masked in the ragged tail iteration). LDS byte address = low 32
    // bits of the flat pointer (LDS aperture maps addr[31:0] -> LDS offset).
    for (int k = (int)threadIdx.x; k < NZ; k += 256) {
        unsigned     lds_off = (unsigned)(uintptr_t)(&zs[k]);
        const int*   gptr    = ZposT + k;
        asm volatile("global_load_async_to_lds_b32 %0, %1, off"
                     :
                     : "v"(lds_off), "v"(gptr)
                     : "memory");
    }
    asm volatile("s_wait_asynccnt 0" ::: "memory");
    __syncthreads();

    int idx = (int)(blockIdx.x * 256u + threadIdx.x);
    if (idx >= total) return;

    unsigned b = (unsigned)idx / (unsigned)N2;   // strength-reduced to mulhi
    unsigned i = (unsigned)idx - b * (unsigned)N2;

    // CSR segment bounds; adjacent lanes read adjacent entries -> coalesced.
    int start = cap_indptr[idx];
    int end   = cap_indptr[idx + 1];

    // Speculative prefetch of this lane's index segment (global_prefetch_b8);
    // OOB-safe: speculative prefetch translation failures are dropped.
    __builtin_prefetch(cap_indices + start, 0, 1);

    int cp = current_player[b];
    int ch = current_hash[b];

    int z_empty = zs[i];
    int place   = z_empty ^ zs[(cp + 1) * N2 + (int)i];

    // D[b][j] = z_empty[j] ^ ZposT[2-cp][j]; XOR-reduce over captured stones.
    int other = (2 - cp) * N2;
    int seg   = 0;
    for (int k = start; k < end; ++k) {
        int j = cap_indices[k];
        seg ^= zs[j] ^ zs[other + j];
    }

    bool keep = (legal_mask[idx] != 0) & (can_capture_any[idx] != 0) & (end > start);
    out[idx]  = ch ^ place ^ (keep ? seg : 0);
}

extern "C" void kernel_launch(void* const* d_in, const int* in_sizes, int n_in,
                              void* d_out, int out_size, void* d_ws, size_t ws_size,
                              hipStream_t stream) {
    const uint8_t* legal      = (const uint8_t*)d_in[0];
    const int*     cur_player = (const int*)d_in[1];
    const int*     cur_hash   = (const int*)d_in[2];
    const int*     ZposT      = (const int*)d_in[3];
    const uint8_t* cca        = (const uint8_t*)d_in[4];
    const int*     indptr     = (const int*)d_in[5];
    const int*     indices    = (const int*)d_in[6];
    int*           out        = (int*)d_out;

    int total  = in_sizes[0];          // B * N2 (flat size of legal_mask)
    int blocks = (total + 255) / 256;

    hipLaunchKernelGGL(superko_hash_kernel, dim3(blocks), dim3(256), 0, stream,
                       legal, cur_player, cur_hash, ZposT, cca, indptr, indices,
                       out, total);
}